// MultiscaleNormalizedCrossCorrelation2d_35313221107869
// MI455X (gfx1250) — compile-verified
//
#include <hip/hip_runtime.h>
#include <math.h>

// MultiscaleNormalizedCrossCorrelation2d for MI455X (gfx1250, wave32).
// Patch (13x13) NCC box sums computed as banded GEMMs on V_WMMA_F32_16X16X4_F32:
//   S(16x16) = Bv(16x28) * Q(28x28) * Bh(28x16),  Bv/Bh are 0/1 band matrices
// generated from lane IDs (zero memory cost). Global NCC sums are fused into the
// same single pass over the inputs (each tile owns a unique 16x16 core).

typedef __attribute__((ext_vector_type(2))) float v2f;
typedef __attribute__((ext_vector_type(8))) float v8f;

#define IMH   512
#define IMW   512
#define PATCH 13
#define HP    500            // IMH - PATCH + 1
#define TILE  16
#define REG   28             // TILE + PATCH - 1
#define KSTEP 7              // REG / 4
#define SXW   30             // padded row stride (keeps pairs 8B-aligned, odd/64 banks)
#define NCC_EPS 1e-5f

// 1.0f if c <= r <= c+12 (13-wide band), else 0.0f
__device__ __forceinline__ float band01(int r, int c) {
    return ((unsigned)(r - c) <= 12u) ? 1.0f : 0.0f;
}

__device__ __forceinline__ v8f wmma4(v2f a, v2f b, v8f c) {
    return __builtin_amdgcn_wmma_f32_16x16x4_f32(false, a, false, b,
                                                 (short)0, c, false, false);
}

__global__ void __launch_bounds__(64) ncc_zero(float* ws, int n) {
    int i = blockIdx.x * blockDim.x + threadIdx.x;
    if (i < n) ws[i] = 0.0f;
}

// grid = (32 tiles_x, 32 tiles_y, B), block = 32 (one wave; WMMA is wave-scoped)
__global__ void __launch_bounds__(32)
ncc_patch(const float* __restrict__ x1, const float* __restrict__ x2,
          float* __restrict__ ws, int B) {
    const int tx = blockIdx.x, ty = blockIdx.y, b = blockIdx.z;
    const int lane = threadIdx.x;
    const int m  = lane & 15;   // M (A rows) / N (B,D cols) index
    const int hi = lane >> 4;   // lane half selects K pair
    const int row0 = ty * TILE, col0 = tx * TILE;
    const float* __restrict__ p1 = x1 + (size_t)b * IMH * IMW;
    const float* __restrict__ p2 = x2 + (size_t)b * IMH * IMW;

    __shared__ float sX1[32][SXW];       // rows 28..31 zero (A-matrix pad rows)
    __shared__ float sX2[32][SXW];
    __shared__ float sT[5][32][TILE + 1]; // T staging for all 5 quantities

    // ---- cooperative load of the 28x28 input region (zero-pad OOB) ----
    for (int idx = lane; idx < REG * REG; idx += 32) {
        int r = idx / REG, c = idx - r * REG;
        int gr = row0 + r, gc = col0 + c;
        bool ok = (gr < IMH) && (gc < IMW);
        size_t off = (size_t)gr * IMW + gc;
        sX1[r][c] = ok ? p1[off] : 0.0f;
        sX2[r][c] = ok ? p2[off] : 0.0f;
    }
    for (int idx = lane; idx < 4 * REG; idx += 32) {   // zero pad rows 28..31
        int r = 28 + idx / REG, c = idx - (idx / REG) * REG;
        sX1[r][c] = 0.0f;
        sX2[r][c] = 0.0f;
    }
    __syncthreads();

    // ---- fused global-NCC partial sums over this tile's unique 16x16 core ----
    {
        float s1 = 0.f, s2 = 0.f, s11 = 0.f, s22 = 0.f, s12 = 0.f;
        for (int idx = lane; idx < TILE * TILE; idx += 32) {
            int r = idx >> 4, c = idx & 15;
            float a = sX1[r][c], bb = sX2[r][c];
            s1 += a; s2 += bb; s11 += a * a; s22 += bb * bb; s12 += a * bb;
        }
        #pragma unroll
        for (int off = 16; off > 0; off >>= 1) {
            s1  += __shfl_xor(s1,  off, 32);
            s2  += __shfl_xor(s2,  off, 32);
            s11 += __shfl_xor(s11, off, 32);
            s22 += __shfl_xor(s22, off, 32);
            s12 += __shfl_xor(s12, off, 32);
        }
        if (lane == 0) {
            float* g = ws + B + (size_t)b * 5;
            atomicAdd(&g[0], s1);  atomicAdd(&g[1], s2);
            atomicAdd(&g[2], s11); atomicAdd(&g[3], s22);
            atomicAdd(&g[4], s12);
        }
    }

    // ---- hoisted band fragments: identical for Bh (B frag) and Bv (A frag) ----
    v2f bandf[KSTEP];
    #pragma unroll
    for (int k = 0; k < KSTEP; ++k) {
        const int c0 = 4 * k + 2 * hi;
        bandf[k].x = band01(c0,     m);
        bandf[k].y = band01(c0 + 1, m);
    }

    // ---- horizontal pass: T_q = Q_q(28x28) * Bh(28x16), all 5 quantities ----
    // 10 independent WMMA accumulation chains (hides D->C RAW latency).
    v8f tT[5], tB[5];
    #pragma unroll
    for (int q = 0; q < 5; ++q) { tT[q] = (v8f){}; tB[q] = (v8f){}; }

    #pragma unroll
    for (int k = 0; k < KSTEP; ++k) {
        const int c0 = 4 * k + 2 * hi;
        // raw values read once per k-step (top rows m, bottom rows 16+m)
        float a0 = sX1[m][c0],      a1 = sX1[m][c0 + 1];
        float b0 = sX2[m][c0],      b1 = sX2[m][c0 + 1];
        float c0v = sX1[16 + m][c0], c1v = sX1[16 + m][c0 + 1];
        float d0 = sX2[16 + m][c0], d1 = sX2[16 + m][c0 + 1];
        const v2f bf = bandf[k];
        tT[0] = wmma4((v2f){a0, a1},           bf, tT[0]);   // x1
        tB[0] = wmma4((v2f){c0v, c1v},         bf, tB[0]);
        tT[1] = wmma4((v2f){b0, b1},           bf, tT[1]);   // x2
        tB[1] = wmma4((v2f){d0, d1},           bf, tB[1]);
        tT[2] = wmma4((v2f){a0 * a0, a1 * a1}, bf, tT[2]);   // x1^2
        tB[2] = wmma4((v2f){c0v * c0v, c1v * c1v}, bf, tB[2]);
        tT[3] = wmma4((v2f){b0 * b0, b1 * b1}, bf, tT[3]);   // x2^2
        tB[3] = wmma4((v2f){d0 * d0, d1 * d1}, bf, tB[3]);
        tT[4] = wmma4((v2f){a0 * b0, a1 * b1}, bf, tT[4]);   // x1*x2
        tB[4] = wmma4((v2f){c0v * d0, c1v * d1}, bf, tB[4]);
    }

    // stage all T matrices (D-layout -> plain [row][col]) behind one barrier
    #pragma unroll
    for (int q = 0; q < 5; ++q) {
        #pragma unroll
        for (int i = 0; i < 8; ++i) {
            sT[q][i + 8 * hi][m]      = tT[q][i];   // T rows 0..15
            sT[q][16 + i + 8 * hi][m] = tB[q][i];   // T rows 16..31 (28..31 = 0)
        }
    }
    __syncthreads();

    // ---- vertical pass: S_q = Bv(16x28) * T_q(28x16) ----
    v8f acc[5];
    #pragma unroll
    for (int q = 0; q < 5; ++q) {
        v8f s = (v8f){};
        #pragma unroll
        for (int k = 0; k < KSTEP; ++k) {
            const int c0 = 4 * k + 2 * hi;
            v2f bT;
            bT.x = sT[q][c0][m];
            bT.y = sT[q][c0 + 1][m];
            s = wmma4(bandf[k], bT, s);
        }
        acc[q] = s;
    }

    // ---- per-window NCC + masked reduction ----
    float ccsum = 0.0f;
    const float inv_n = 1.0f / (float)(PATCH * PATCH);
    #pragma unroll
    for (int i = 0; i < 8; ++i) {
        float S1 = acc[0][i], S2 = acc[1][i];
        float S11 = acc[2][i], S22 = acc[3][i], S12 = acc[4][i];
        float m1 = S1 * inv_n, m2 = S2 * inv_n;
        float v1 = S11 * inv_n - m1 * m1;
        float v2 = S22 * inv_n - m2 * m2;
        float cross = S12 - (float)(PATCH * PATCH) * m1 * m2;
        float cc = cross * rsqrtf((v1 + NCC_EPS) * (v2 + NCC_EPS));
        int yo = row0 + i + 8 * hi;                 // D row = i + 8*hi
        int xo = col0 + m;                          // D col = lane&15
        if (yo < HP && xo < HP) ccsum += cc;
    }
    #pragma unroll
    for (int off = 16; off > 0; off >>= 1) ccsum += __shfl_xor(ccsum, off, 32);
    if (lane == 0) atomicAdd(&ws[b], ccsum);
}

__global__ void __launch_bounds__(64)
ncc_final(const float* __restrict__ ws, float* __restrict__ out, int B) {
    int b = blockIdx.x * blockDim.x + threadIdx.x;
    if (b >= B) return;
    const float* g = ws + B + (size_t)b * 5;
    float S1 = g[0], S2 = g[1], S11 = g[2], S22 = g[3], S12 = g[4];
    const float N = (float)(IMH * IMW);
    float m1 = S1 / N, m2 = S2 / N;
    float v1 = S11 / N - m1 * m1;
    float v2 = S22 / N - m2 * m2;
    float dot = (S12 - N * m1 * m2) * rsqrtf((v1 + NCC_EPS) * (v2 + NCC_EPS));
    float gscore = dot / N;
    float pscore = ws[b] / ((float)HP * (float)HP * (float)(PATCH * PATCH));
    out[b] = 0.5f * gscore + 0.5f * pscore;
}

extern "C" void kernel_launch(void* const* d_in, const int* in_sizes, int n_in,
                              void* d_out, int out_size, void* d_ws, size_t ws_size,
                              hipStream_t stream) {
    const float* x1 = (const float*)d_in[0];
    const float* x2 = (const float*)d_in[1];
    float* out = (float*)d_out;
    float* ws  = (float*)d_ws;
    const int B = in_sizes[0] / (IMH * IMW);
    const int nT = (HP + TILE - 1) / TILE;          // 32 tiles per axis

    const int nws = B * 6;                          // [0,B): patch acc, [B,6B): global sums
    ncc_zero<<<dim3((nws + 63) / 64), 64, 0, stream>>>(ws, nws);
    ncc_patch<<<dim3(nT, nT, B), 32, 0, stream>>>(x1, x2, ws, B);
    ncc_final<<<dim3((B + 63) / 64), 64, 0, stream>>>(ws, out, B);
}